// SpatialTransformer3D_89464168776025
// MI455X (gfx1250) — compile-verified
//
#include <hip/hip_runtime.h>
#include <hip/hip_bf16.h>

typedef __attribute__((ext_vector_type(2))) float v2f;
typedef __attribute__((ext_vector_type(8))) float v8f;

#define BATCH 4
#define HH 160
#define WW 160
#define DD 160
#define NVOX (HH * WW * DD)

// Order-preserving float <-> uint key (monotonic: smaller float -> smaller key)
__device__ __forceinline__ unsigned fkey(float f) {
  unsigned u = __float_as_uint(f);
  return u ^ ((unsigned)((int)u >> 31) | 0x80000000u);
}
__device__ __forceinline__ float fdec(unsigned k) {
  unsigned u = (k & 0x80000000u) ? (k ^ 0x80000000u) : ~k;
  return __uint_as_float(u);
}

__global__ void st3d_init_min(unsigned* __restrict__ keys) {
  if (threadIdx.x < BATCH) keys[threadIdx.x] = 0xFFFFFFFFu;
}

__global__ __launch_bounds__(256) void st3d_min_reduce(const float* __restrict__ img,
                                                       unsigned* __restrict__ keys) {
  __shared__ unsigned sm[256];
  const int b = blockIdx.y;
  const float4* p = (const float4*)(img + (size_t)b * NVOX);
  float4 v = p[(size_t)blockIdx.x * 256 + threadIdx.x];
  float m = fminf(fminf(v.x, v.y), fminf(v.z, v.w));
  sm[threadIdx.x] = fkey(m);
  __syncthreads();
  for (int s = 128; s > 0; s >>= 1) {
    if (threadIdx.x < s) {
      unsigned a = sm[threadIdx.x], c = sm[threadIdx.x + s];
      sm[threadIdx.x] = a < c ? a : c;
    }
    __syncthreads();
  }
  if (threadIdx.x == 0) atomicMin(&keys[b], sm[0]);
}

__global__ __launch_bounds__(256) void st3d_sample(const float* __restrict__ img,
                                                   const float* __restrict__ tf,
                                                   const unsigned* __restrict__ keys,
                                                   float* __restrict__ out) {
  __shared__ float sth[12];  // theta: 3 rows x 4 cols (R | t)

  const int tid  = threadIdx.x;
  const int lane = tid & 31;
  const int wv   = tid >> 5;
  const int b    = blockIdx.z / HH;
  const int h    = blockIdx.z % HH;
  const int w    = blockIdx.y * 8 + wv;
  const int db   = blockIdx.x * 32;
  const int d    = db + lane;

  if (tid == 0) {
    const float qx = tf[b * 7 + 0], qy = tf[b * 7 + 1], qz = tf[b * 7 + 2], qw = tf[b * 7 + 3];
    const float tx = 2.f * qx, ty = 2.f * qy, tz = 2.f * qz;
    const float twx = tx * qw, twy = ty * qw, twz = tz * qw;
    const float txx = tx * qx, txy = ty * qx, txz = tz * qx;
    const float tyy = ty * qy, tyz = tz * qy, tzz = tz * qz;
    sth[0] = 1.f - (tyy + tzz); sth[1] = txy - twz;         sth[2]  = txz + twy;         sth[3]  = tf[b * 7 + 4];
    sth[4] = txy + twz;         sth[5] = 1.f - (txx + tzz); sth[6]  = tyz - twx;         sth[7]  = tf[b * 7 + 5];
    sth[8] = txz - twy;         sth[9] = tyz + twx;         sth[10] = 1.f - (txx + tyy); sth[11] = tf[b * 7 + 6];
  }
  __syncthreads();

  // ---- pts = theta (3x4) @ grid (4x16) via V_WMMA_F32_16X16X4_F32 ----
  // A layout (16x4 f32): lane m%16 = row M; VGPR v holds K = v + 2*(lane>=16).
  // B layout (4x16 f32): lane n%16 = col N; VGPR v holds K = v + 2*(lane>=16).
  // D layout (16x16 f32): VGPR r, lanes 0-15 -> row r; lanes 16-31 -> row r+8.
  const int half = lane >> 4;
  const int n16  = lane & 15;
  const int k0   = half * 2;

  float a0x = 0.f, a0y = 0.f, a1x = 0.f, a1y = 0.f;
  if (n16 < 3) {                      // WMMA0: theta in rows 0..2 -> lanes 0..15 read rows 0..2
    a0x = sth[n16 * 4 + k0];
    a0y = sth[n16 * 4 + k0 + 1];
  }
  if (n16 >= 8 && n16 < 11) {         // WMMA1: theta in rows 8..10 -> lanes 16..31 read rows 8..10
    a1x = sth[(n16 - 8) * 4 + k0];
    a1y = sth[(n16 - 8) * 4 + k0 + 1];
  }

  // Grid columns: rows K = (x, y, z, 1). x,y are wave-uniform; z varies with point index.
  const float xw  = -1.f + (float)w * (2.f / (float)(WW - 1));
  const float yh  = -1.f + (float)h * (2.f / (float)(HH - 1));
  const float zn0 = -1.f + (float)(db + n16)      * (2.f / (float)(DD - 1));
  const float zn1 = -1.f + (float)(db + 16 + n16) * (2.f / (float)(DD - 1));

  v2f a0; a0.x = a0x; a0.y = a0y;
  v2f a1; a1.x = a1x; a1.y = a1y;
  v2f b0; b0.x = half ? zn0 : xw; b0.y = half ? 1.f : yh;
  v2f b1; b1.x = half ? zn1 : xw; b1.y = half ? 1.f : yh;
  v8f c = {};
  v8f d0 = __builtin_amdgcn_wmma_f32_16x16x4_f32(false, a0, false, b0, (short)0, c, false, false);
  v8f d1 = __builtin_amdgcn_wmma_f32_16x16x4_f32(false, a1, false, b1, (short)0, c, false, false);

  const float px = half ? d1[0] : d0[0];
  const float py = half ? d1[1] : d0[1];
  const float pz = half ? d1[2] : d0[2];

  // ---- trilinear sampling (clip semantics identical to reference) ----
  const float cw = (float)(WW - 1) * 0.5f;
  const float ch = (float)(HH - 1) * 0.5f;
  const float cd = (float)(DD - 1) * 0.5f;
  const float fx = cw * px + cw;
  const float fy = ch * py + ch;
  const float fz = cd * pz + cd;

  const bool valid = (fx >= 0.f) && (fx <= (float)(WW - 1)) &&
                     (fy >= 0.f) && (fy <= (float)(HH - 1)) &&
                     (fz >= 0.f) && (fz <= (float)(DD - 1));

  const float x0f = fminf(fmaxf(floorf(fx), 0.f), (float)(WW - 1));
  const float y0f = fminf(fmaxf(floorf(fy), 0.f), (float)(HH - 1));
  const float z0f = fminf(fmaxf(floorf(fz), 0.f), (float)(DD - 1));

  const float wx0 = fminf(fmaxf(1.f - fabsf(fx - x0f), 0.f), 1.f);
  const float wx1 = fminf(fmaxf(1.f - fabsf(fx - (x0f + 1.f)), 0.f), 1.f);
  const float wy0 = fminf(fmaxf(1.f - fabsf(fy - y0f), 0.f), 1.f);
  const float wy1 = fminf(fmaxf(1.f - fabsf(fy - (y0f + 1.f)), 0.f), 1.f);
  const float wz0 = fminf(fmaxf(1.f - fabsf(fz - z0f), 0.f), 1.f);
  const float wz1 = fminf(fmaxf(1.f - fabsf(fz - (z0f + 1.f)), 0.f), 1.f);

  const int ix0 = (int)x0f, ix1 = min(ix0 + 1, WW - 1);
  const int iy0 = (int)y0f, iy1 = min(iy0 + 1, HH - 1);
  const int iz0 = (int)z0f, iz1 = min(iz0 + 1, DD - 1);

  const float* base = img + (size_t)b * NVOX;   // image indexed as [y][x][z]
  const int oy0 = iy0 * (WW * DD), oy1 = iy1 * (WW * DD);
  const int ox0 = ix0 * DD,        ox1 = ix1 * DD;

  const float acc =
      wx0 * wy0 * wz0 * base[oy0 + ox0 + iz0] +
      wx0 * wy0 * wz1 * base[oy0 + ox0 + iz1] +
      wx0 * wy1 * wz0 * base[oy1 + ox0 + iz0] +
      wx0 * wy1 * wz1 * base[oy1 + ox0 + iz1] +
      wx1 * wy0 * wz0 * base[oy0 + ox1 + iz0] +
      wx1 * wy0 * wz1 * base[oy0 + ox1 + iz1] +
      wx1 * wy1 * wz0 * base[oy1 + ox1 + iz0] +
      wx1 * wy1 * wz1 * base[oy1 + ox1 + iz1];

  const float fill = fdec(keys[b]);
  out[(((size_t)b * HH + h) * WW + w) * DD + d] = valid ? acc : fill;
}

extern "C" void kernel_launch(void* const* d_in, const int* in_sizes, int n_in,
                              void* d_out, int out_size, void* d_ws, size_t ws_size,
                              hipStream_t stream) {
  const float* img = (const float*)d_in[0];   // (4,160,160,160,1) f32
  const float* tf  = (const float*)d_in[1];   // (4,7) f32
  float* out       = (float*)d_out;           // (4,160,160,160,1) f32
  unsigned* keys   = (unsigned*)d_ws;         // 4 per-batch min keys

  st3d_init_min<<<1, 32, 0, stream>>>(keys);
  st3d_min_reduce<<<dim3(NVOX / 1024, BATCH, 1), 256, 0, stream>>>(img, keys);
  // grid: x = D/32 (lanes along d), y = W/8 (one w per wave), z = B*H
  st3d_sample<<<dim3(DD / 32, WW / 8, BATCH * HH), 256, 0, stream>>>(img, tf, keys, out);
}